// JPEG_11802570129526
// MI455X (gfx1250) — compile-verified
//
#include <hip/hip_runtime.h>

// fp32 WMMA fragments (wave32): A/B 16x4 -> v2f per lane, C/D 16x16 -> v8f
typedef __attribute__((ext_vector_type(2))) float v2f;
typedef __attribute__((ext_vector_type(4))) float v4f;
typedef __attribute__((ext_vector_type(8))) float v8f;

// Orthonormal DCT-II matrix C (8x8), exact values:
//  s = 0.5*cos(pi*m/16):  s1=0.4903926402, s2=0.4619397663, s3=0.4157348062,
//  s4=0.3535533906, s5=0.2777851165, s6=0.1913417162, s7=0.0975451610
// M = blockdiag(C, C) as a full 16x16 constant table (zeros off-diagonal).
#define S0 0.35355339059327373f
#define S1 0.49039264020161522f
#define S2 0.46193976625564337f
#define S3 0.41573480615127262f
#define S4 0.35355339059327379f
#define S5 0.27778511650980114f
#define S6 0.19134171618254492f
#define S7 0.09754516100806413f
#define Z8 0.f,0.f,0.f,0.f,0.f,0.f,0.f,0.f
__device__ __constant__ float kM[16][16] = {
 { S0, S0, S0, S0, S0, S0, S0, S0, Z8},
 { S1, S3, S5, S7,-S7,-S5,-S3,-S1, Z8},
 { S2, S6,-S6,-S2,-S2,-S6, S6, S2, Z8},
 { S3,-S7,-S1,-S5, S5, S1, S7,-S3, Z8},
 { S4,-S4,-S4, S4, S4,-S4,-S4, S4, Z8},
 { S5,-S1, S7, S3,-S3,-S7, S1,-S5, Z8},
 { S6,-S2, S2,-S6,-S6, S2,-S2, S6, Z8},
 { S7,-S5, S3,-S1, S1,-S3, S5,-S7, Z8},
 {Z8,  S0, S0, S0, S0, S0, S0, S0, S0},
 {Z8,  S1, S3, S5, S7,-S7,-S5,-S3,-S1},
 {Z8,  S2, S6,-S6,-S2,-S2,-S6, S6, S2},
 {Z8,  S3,-S7,-S1,-S5, S5, S1, S7,-S3},
 {Z8,  S4,-S4,-S4, S4, S4,-S4,-S4, S4},
 {Z8,  S5,-S1, S7, S3,-S3,-S7, S1,-S5},
 {Z8,  S6,-S2, S2,-S6,-S6, S2,-S2, S6},
 {Z8,  S7,-S5, S3,-S1, S1,-S3, S5,-S7}};

// 100 / LUMINANCE_Q  (reciprocal quantization table)
__device__ __constant__ float kQinv[64] = {
 6.25f,      9.0909091f, 10.0f,      6.25f,      4.1666667f, 2.5f,       1.9607843f, 1.6393443f,
 8.3333333f, 8.3333333f, 7.1428571f, 5.2631579f, 3.8461538f, 1.7241379f, 1.6666667f, 1.8181818f,
 7.1428571f, 7.6923077f, 6.25f,      4.1666667f, 2.5f,       1.7543860f, 1.4492754f, 1.7857143f,
 7.1428571f, 5.8823529f, 4.5454545f, 3.4482759f, 1.9607843f, 1.1494253f, 1.25f,      1.6129032f,
 5.5555556f, 4.5454545f, 2.7027027f, 1.7857143f, 1.4705882f, 0.91743119f,0.97087379f,1.2987013f,
 4.1666667f, 2.7777778f, 1.8181818f, 1.5625f,    1.2345679f, 0.96153846f,0.88495575f,1.0869565f,
 2.0408163f, 1.5625f,    1.2820513f, 1.1494253f, 0.97087379f,0.82644628f,0.83333333f,0.99009901f,
 1.3888889f, 1.0869565f, 1.0526316f, 1.0204082f, 0.89285714f,1.0f,       0.97087379f,1.0101010f};

__global__ __launch_bounds__(256) void jpeg_dct8x8_wmma(
        const float* __restrict__ img, const float* __restrict__ qf,
        float* __restrict__ out)
{
    constexpr int W = 1024, H = 1024;
    constexpr int TX = W / 16, TY = H / 16;   // 64 x 64 tiles of 16x16
    constexpr int BX = W / 8,  BY = H / 8;    // 128 x 128 JPEG blocks

    const int lane = threadIdx.x & 31;
    const int wave = threadIdx.x >> 5;
    const int tile = blockIdx.x * 8 + wave;   // one wave = one 16x16 tile (4 blocks)
    const int tx = tile % TX;
    const int ty = (tile / TX) % TY;
    const int b  = tile / (TX * TY);
    if (b >= 16) return;                      // wave-uniform guard

    const int hi = lane >> 4;                 // lane group (0: lanes 0-15, 1: 16-31)
    const int ln = lane & 15;

    // ---- A fragments of M from constant table: col pair (4kc+2hi, +1) is one b64 ----
    const v2f* mrow = (const v2f*)(&kM[ln][0]);
    v2f a[4];
#pragma unroll
    for (int kc = 0; kc < 4; ++kc)
        a[kc] = mrow[2 * kc + hi];

    // ---- B fragments of X (16x16 pixel tile), centered by -128 ----
    const float* base = img + ((size_t)b * H + (size_t)ty * 16) * W + (size_t)tx * 16;
    v2f xb[4];
#pragma unroll
    for (int kc = 0; kc < 4; ++kc) {
        const int r = 4 * kc + 2 * hi;        // B 4x16 layout: K = e + 2*hi
        xb[kc][0] = base[(size_t)(r + 0) * W + ln] - 128.0f;
        xb[kc][1] = base[(size_t)(r + 1) * W + ln] - 128.0f;
    }

    // ---- Y = M * X : 4 chained v_wmma_f32_16x16x4_f32 ----
    v8f y = {};
#pragma unroll
    for (int kc = 0; kc < 4; ++kc)
        y = __builtin_amdgcn_wmma_f32_16x16x4_f32(false, a[kc], false, xb[kc],
                                                  (short)0, y, false, false);

    // ---- Transpose Y through per-wave LDS (stride 17 to spread banks) ----
    __shared__ float lds[8][16 * 17];
    float* t = lds[wave];
#pragma unroll
    for (int v = 0; v < 8; ++v)               // C/D layout: lane holds Y[v+8*hi][ln]
        t[(v + 8 * hi) * 17 + ln] = y[v];
    // cross-lane LDS RAW within the wave: compiler can't track it -> explicit wait
    asm volatile("s_wait_dscnt 0" ::: "memory");

    v2f yb[4];                                // B fragments of Y^T
#pragma unroll
    for (int kc = 0; kc < 4; ++kc) {
        const int c0 = 4 * kc + 2 * hi;
        yb[kc][0] = t[ln * 17 + c0 + 0];      // Y^T[k][n] = Y[n][k]
        yb[kc][1] = t[ln * 17 + c0 + 1];
    }

    // ---- D^T = M * Y^T  (D = C X C^T per 8x8 quadrant) ----
    v8f d = {};
#pragma unroll
    for (int kc = 0; kc < 4; ++kc)
        d = __builtin_amdgcn_wmma_f32_16x16x4_f32(false, a[kc], false, yb[kc],
                                                  (short)0, d, false, false);

    // ---- Quantization scale: one wave-uniform reciprocal + constant 1/Q table ----
    const float q = qf[b];
    const float factor = (q < 50.0f) ? (5000.0f / q) : (200.0f - 2.0f * q);
    const float rfac = 1.0f / factor;

    // lane holds D^T[v+8*hi][ln] = D[ln][v+8*hi] :  u = ln&7, qr = ln>>3, vv = v, qc = hi
    const int u  = ln & 7;
    const v4f qi0 = ((const v4f*)(&kQinv[u * 8]))[0];
    const v4f qi1 = ((const v4f*)(&kQinv[u * 8]))[1];

    const int by = ty * 2 + (ln >> 3);
    const int bx = tx * 2 + hi;
    float* obase = out + (((size_t)b * 64 + (size_t)u * 8) * BY + by) * BX + bx;
#pragma unroll
    for (int v = 0; v < 8; ++v) {
        const float qinv = (v < 4) ? qi0[v & 3] : qi1[v & 3];
        obase[(size_t)v * BY * BX] = d[v] * (rfac * qinv);  // channel stride = 128*128
    }
}

extern "C" void kernel_launch(void* const* d_in, const int* in_sizes, int n_in,
                              void* d_out, int out_size, void* d_ws, size_t ws_size,
                              hipStream_t stream) {
    const float* img = (const float*)d_in[0];   // [16,1,1024,1024] f32
    const float* qf  = (const float*)d_in[1];   // [16] f32
    float* out = (float*)d_out;                 // [16,64,128,128] f32
    (void)in_sizes; (void)n_in; (void)out_size; (void)d_ws; (void)ws_size;
    // 16*64*64 = 65536 tiles, 8 waves (tiles) per 256-thread block
    jpeg_dct8x8_wmma<<<dim3(65536 / 8), dim3(256), 0, stream>>>(img, qf, out);
}